// SAGE_67551245631656
// MI455X (gfx1250) — compile-verified
//
#include <hip/hip_runtime.h>

#define F_DIM 128
#define D1_DIM 64

typedef __attribute__((ext_vector_type(2))) float v2f;
typedef __attribute__((ext_vector_type(8))) float v8f;

// ---------------- K1: deg = 1.0 (self loop), s = 0, r64 = 0 ----------------
__global__ void gcn_init_kernel(float* __restrict__ deg, float* __restrict__ s,
                                float* __restrict__ r64, int n) {
    int i = blockIdx.x * blockDim.x + threadIdx.x;
    if (i < n) { deg[i] = 1.0f; s[i] = 0.0f; }
    if (i < D1_DIM) r64[i] = 0.0f;
}

// ---------------- K2: deg[dst] += 1 ----------------
__global__ void gcn_deg_kernel(const int* __restrict__ dst, float* __restrict__ deg, int e) {
    int i = blockIdx.x * blockDim.x + threadIdx.x;
    if (i < e) atomicAdd(&deg[dst[i]], 1.0f);
}

// ---------------- K3: dis = rsqrt(deg) in place (deg >= 1 always) ----------------
__global__ void gcn_rsqrt_kernel(float* __restrict__ deg, int n) {
    int i = blockIdx.x * blockDim.x + threadIdx.x;
    if (i < n) deg[i] = rsqrtf(deg[i]);
}

// ---------------- K4: s[src] += dis[dst] ----------------
__global__ void gcn_srcsum_kernel(const int* __restrict__ src, const int* __restrict__ dst,
                                  const float* __restrict__ dis, float* __restrict__ s, int e) {
    int i = blockIdx.x * blockDim.x + threadIdx.x;
    if (i < e) atomicAdd(&s[src[i]], dis[dst[i]]);
}

// ---------------- K5: c = dis * (s + dis), stored into s in place ----------------
__global__ void gcn_coef_kernel(const float* __restrict__ dis, float* __restrict__ s, int n) {
    int i = blockIdx.x * blockDim.x + threadIdx.x;
    if (i < n) {
        float d = dis[i];
        s[i] = d * (s[i] + d);
    }
}

// Per-wave 16-row stripe: 16x64 output tile via V_WMMA_F32_16X16X4_F32, then
// weighted row-reduction (weights c[row]) into ldsR[64].
// ldsP layout: v2f ldsP[q*64 + col] = { W[2q, col], W[2q+1, col] } so every
// B fragment is one aligned ds_load_b64 (no VGPR repacking).
template <bool TAIL>
__device__ __forceinline__ void gcn_stripe(
    const float* __restrict__ X, const v2f* __restrict__ ldsP,
    const float* __restrict__ c, float* __restrict__ ldsR,
    int r0, int lane, int n)
{
    const int hi = lane >> 4;          // 0: K pair {k,k+1}; 1: K pair {k+2,k+3}
    const int lm = lane & 15;
    const int row = r0 + lm;
    const int rowc = (!TAIL || row < n) ? row : n - 1;
    const float mask = (!TAIL || row < n) ? 1.0f : 0.0f;
    const float* __restrict__ xrow = X + (size_t)rowc * F_DIM;

    // Pull the lane's whole 512B feature row into cache up front
    // (k-loop otherwise walks it 8B at a time). -> global_prefetch_b8
    __builtin_prefetch(xrow, 0, 1);
    __builtin_prefetch(xrow + 32, 0, 1);
    __builtin_prefetch(xrow + 64, 0, 1);
    __builtin_prefetch(xrow + 96, 0, 1);

    v8f acc0 = {}, acc1 = {}, acc2 = {}, acc3 = {};

    #pragma unroll 4
    for (int k = 0; k < F_DIM; k += 4) {
        const int ka = k + 2 * hi;                 // even
        v2f a = *(const v2f*)(xrow + ka);          // A[m, ka], A[m, ka+1]
        if (TAIL) { a.x *= mask; a.y *= mask; }

        const int base = (ka >> 1) * D1_DIM + lm;  // (k/2 + hi)*64 + lm
        v2f b0  = ldsP[base];
        v2f b1v = ldsP[base + 16];
        v2f b2  = ldsP[base + 32];
        v2f b3  = ldsP[base + 48];

        acc0 = __builtin_amdgcn_wmma_f32_16x16x4_f32(false, a, false, b0,  (short)0, acc0, false, false);
        acc1 = __builtin_amdgcn_wmma_f32_16x16x4_f32(false, a, false, b1v, (short)0, acc1, false, false);
        acc2 = __builtin_amdgcn_wmma_f32_16x16x4_f32(false, a, false, b2,  (short)0, acc2, false, false);
        acc3 = __builtin_amdgcn_wmma_f32_16x16x4_f32(false, a, false, b3,  (short)0, acc3, false, false);
    }

    // D layout: VGPR v -> M = v + 8*hi, N = lane&15.
    float cw[8];
    #pragma unroll
    for (int v = 0; v < 8; ++v) {
        const int rr = r0 + v + 8 * hi;
        cw[v] = (!TAIL || rr < n) ? c[rr] : 0.0f;
    }
    float p0 = 0.f, p1 = 0.f, p2 = 0.f, p3 = 0.f;
    #pragma unroll
    for (int v = 0; v < 8; ++v) {
        p0 += cw[v] * acc0[v];
        p1 += cw[v] * acc1[v];
        p2 += cw[v] * acc2[v];
        p3 += cw[v] * acc3[v];
    }
    atomicAdd(&ldsR[lm],      p0);     // ds_add_f32
    atomicAdd(&ldsR[16 + lm], p1);
    atomicAdd(&ldsR[32 + lm], p2);
    atomicAdd(&ldsR[48 + lm], p3);
}

// ---------------- K6: fused WMMA GEMM (X @ W1) + weighted column reduction ----------
__global__ __launch_bounds__(256) void gcn_wmma_reduce_kernel(
    const float* __restrict__ X,      // [n, 128]
    const float* __restrict__ W,      // [128, 64]
    const float* __restrict__ c,      // [n]
    float* __restrict__ r64,          // [64]
    int n)
{
    __shared__ v2f   ldsP[(F_DIM / 2) * D1_DIM];  // 32 KB: W1 pre-paired along K
    __shared__ float ldsR[D1_DIM];

    const int tid = threadIdx.x;
    for (int p = tid; p < (F_DIM / 2) * D1_DIM; p += 256) {
        const int q   = p >> 6;        // K pair index
        const int col = p & 63;
        v2f v;
        v.x = W[(2 * q) * D1_DIM + col];
        v.y = W[(2 * q + 1) * D1_DIM + col];
        ldsP[p] = v;
    }
    if (tid < D1_DIM) ldsR[tid] = 0.0f;
    __syncthreads();

    const int wave = tid >> 5;         // 8 waves/block -> 128 rows/block
    const int lane = tid & 31;
    const int r0   = blockIdx.x * 128 + wave * 16;

    if (r0 + 16 <= n) {                // full stripe: no masking anywhere
        gcn_stripe<false>(X, ldsP, c, ldsR, r0, lane, n);
    } else if (r0 < n) {               // generic tail (unused when n % 16 == 0)
        gcn_stripe<true>(X, ldsP, c, ldsR, r0, lane, n);
    }
    __syncthreads();
    if (tid < D1_DIM) atomicAdd(&r64[tid], ldsR[tid]);
}

// ---------------- K7: out[d] = (r64[d] + N*b1[d]) / 16 ----------------
__global__ void gcn_final_kernel(const float* __restrict__ r64, const float* __restrict__ b1,
                                 float* __restrict__ out, int n) {
    int d = threadIdx.x;
    if (d < D1_DIM) out[d] = (r64[d] + (float)n * b1[d]) * (1.0f / 16.0f);
}

extern "C" void kernel_launch(void* const* d_in, const int* in_sizes, int n_in,
                              void* d_out, int out_size, void* d_ws, size_t ws_size,
                              hipStream_t stream) {
    const float* feats = (const float*)d_in[0];   // [N,128]
    const int*   edges = (const int*)d_in[1];     // [2,E]
    const float* W1    = (const float*)d_in[2];   // [128,64]
    const float* b1    = (const float*)d_in[3];   // [64]
    // d_in[4..7] (fc1/fc2) are mathematically dead: softmax rows sum to 1,
    // so mean(assign.T @ nf2, axis=0) == colsum(nf2)/16 exactly.

    const int n = in_sizes[0] / F_DIM;
    const int e = in_sizes[1] / 2;
    const int* src = edges;
    const int* dst = edges + e;

    float* dis = (float*)d_ws;        // n floats (deg, then rsqrt in place)
    float* s   = dis + n;             // n floats (sum dis[dst], then c in place)
    float* r64 = s + n;               // 64 floats

    const int tb = 256;
    gcn_init_kernel  <<<(n + tb - 1) / tb, tb, 0, stream>>>(dis, s, r64, n);
    gcn_deg_kernel   <<<(e + tb - 1) / tb, tb, 0, stream>>>(dst, dis, e);
    gcn_rsqrt_kernel <<<(n + tb - 1) / tb, tb, 0, stream>>>(dis, n);
    gcn_srcsum_kernel<<<(e + tb - 1) / tb, tb, 0, stream>>>(src, dst, dis, s, e);
    gcn_coef_kernel  <<<(n + tb - 1) / tb, tb, 0, stream>>>(dis, s, n);
    gcn_wmma_reduce_kernel<<<(n + 127) / 128, 256, 0, stream>>>(feats, W1, s, r64, n);
    gcn_final_kernel <<<1, D1_DIM, 0, stream>>>(r64, b1, (float*)d_out, n);
}